// AttentionInterpolator_7413113552896
// MI455X (gfx1250) — compile-verified
//
#include <hip/hip_runtime.h>
#include <math.h>

// ---------------------------------------------------------------------------
// Sizes from the reference
//   M=12 members, T_SOL=64, NX=NY=256, N_TIMES=50, NOUT=256
//   out = [2, 50, 256, 256] f32
// ---------------------------------------------------------------------------
#define MM 12
#define TSOL 64
#define NG 256
#define NT 50

typedef __attribute__((ext_vector_type(2))) float v2f;
typedef __attribute__((ext_vector_type(8))) float v8f;

#if defined(__gfx1250__) && __has_builtin(__builtin_amdgcn_wmma_f32_16x16x4_f32)
#define HAVE_WMMA_F32X4 1
#endif

// Member-m resample grid value, replicating JAX f32 rounding:
//   u = linspace(0,1,256) (endpoint exact), ys = (u * lys) * (75/lys)
__device__ __forceinline__ float ys_val(int k, float lys, float s) {
    float u = (k == 255) ? 1.0f : (float)k * (1.0f / 255.0f);
    return (u * lys) * s;
}

// ---------------------------------------------------------------------------
// Kernel A: one wave. Computes the 12 blend weights.
//   K = p_norm(12x3, zero-padded to 16x4) @ Wk^T (two 4x16 tiles) + bk
//   done with V_WMMA_F32_16X16X4_F32 (bias in the C operand, exact f32).
//   Softmax / gaussian / normalization done serially on lane 0 (12 elements).
// ---------------------------------------------------------------------------
__global__ __launch_bounds__(32)
void weights_kernel(const float* __restrict__ params,
                    const float* __restrict__ Wq, const float* __restrict__ bq,
                    const float* __restrict__ Wk, const float* __restrict__ bk,
                    float* __restrict__ w_out) {
    __shared__ float Kmat[16][32];
    __shared__ float qv[32];

    const int lane = threadIdx.x;             // 0..31
    // target point, computed in double then cast (matches python scalars -> f32)
    const float T0 = (float)((75.0 - 30.0) / 90.0);
    const float T1 = (float)(0.001 / 0.0029);
    const float T2 = (float)(0.0001 / 0.0018);

    // ---- normalized params row for this lane's A-row (branchless, masked) ----
    const int row  = lane & 15;
    const int rsel = (row < MM) ? row : 0;
    const float msk = (row < MM) ? 1.0f : 0.0f;
    float p0 = ((params[rsel * 3 + 0] - 30.0f) / 90.0f) * msk;
    float p1 = ( params[rsel * 3 + 1] / 0.0029f)        * msk;
    float p2 = ( params[rsel * 3 + 2] / 0.0018f)        * msk;

    const int  ncol = lane & 15;
    const bool hi   = (lane >= 16);

#ifdef HAVE_WMMA_F32X4
    // A 16x4 layout: lanes 0-15 {K=0,K=1}, lanes 16-31 {K=2,K=3(pad 0)}
    v2f a;
    a.x = hi ? p2 : p0;
    a.y = hi ? 0.0f : p1;

    // B 4x16 tiles (B[k][n] = Wk[n][k]): VGPR0 = rows K0(lo)/K2(hi), VGPR1 = K1/K3
    v2f b0, b1;
    {
        int n0 = ncol, n1 = 16 + ncol;
        b0.x = hi ? Wk[n0 * 3 + 2] : Wk[n0 * 3 + 0];
        b0.y = hi ? 0.0f           : Wk[n0 * 3 + 1];
        b1.x = hi ? Wk[n1 * 3 + 2] : Wk[n1 * 3 + 0];
        b1.y = hi ? 0.0f           : Wk[n1 * 3 + 1];
    }
    // C[M][N] = bk[tile*16+N] (bias, independent of M) -> all 8 comps equal
    v8f c0, c1;
    float bk0 = bk[ncol], bk1 = bk[16 + ncol];
#pragma unroll
    for (int v = 0; v < 8; ++v) { c0[v] = bk0; c1[v] = bk1; }

    v8f d0 = __builtin_amdgcn_wmma_f32_16x16x4_f32(false, a, false, b0,
                                                   (short)0, c0, false, false);
    v8f d1 = __builtin_amdgcn_wmma_f32_16x16x4_f32(false, a, false, b1,
                                                   (short)0, c1, false, false);
    // D layout: comp v, lanes 0-15 -> (M=v, N=lane); lanes 16-31 -> (M=v+8, N=lane-16)
    const int mbase = hi ? 8 : 0;
#pragma unroll
    for (int v = 0; v < 8; ++v) {
        Kmat[mbase + v][ncol]      = d0[v];
        Kmat[mbase + v][16 + ncol] = d1[v];
    }
#else
    // VALU fallback: lane n computes column n of K for all 12 members
    for (int m = 0; m < MM; ++m) {
        float q0 = (params[m * 3 + 0] - 30.0f) / 90.0f;
        float q1 =  params[m * 3 + 1] / 0.0029f;
        float q2 =  params[m * 3 + 2] / 0.0018f;
        Kmat[m][lane] = q0 * Wk[lane * 3 + 0] + q1 * Wk[lane * 3 + 1] +
                        q2 * Wk[lane * 3 + 2] + bk[lane];
    }
    (void)p0; (void)p1; (void)p2; (void)hi; (void)ncol;
#endif

    // Q = t @ Wq^T + bq : one feature per lane
    qv[lane] = T0 * Wq[lane * 3 + 0] + T1 * Wq[lane * 3 + 1] +
               T2 * Wq[lane * 3 + 2] + bq[lane];
    __syncthreads();

    if (lane == 0) {
        const float inv_sqrt8 = 2.8284271247461903f; // divide (as jnp does)
        float attn[MM][4];
        for (int m = 0; m < MM; ++m)
            for (int h = 0; h < 4; ++h) {
                float s = 0.0f;
                for (int d = 0; d < 8; ++d) s += Kmat[m][h * 8 + d] * qv[h * 8 + d];
                attn[m][h] = s / inv_sqrt8;
            }
        // softmax over members (axis 0) per head, then mean over heads
        float attn_w[MM];
        for (int m = 0; m < MM; ++m) attn_w[m] = 0.0f;
        for (int h = 0; h < 4; ++h) {
            float mx = attn[0][h];
            for (int m = 1; m < MM; ++m) mx = fmaxf(mx, attn[m][h]);
            float e[MM], se = 0.0f;
            for (int m = 0; m < MM; ++m) { e[m] = expf(attn[m][h] - mx); se += e[m]; }
            for (int m = 0; m < MM; ++m) attn_w[m] += e[m] / se;
        }
        for (int m = 0; m < MM; ++m) attn_w[m] *= 0.25f;
        // gaussian selection weights
        float sw[MM], ssw = 0.0f;
        for (int m = 0; m < MM; ++m) {
            float q0 = (params[m * 3 + 0] - 30.0f) / 90.0f;
            float q1 =  params[m * 3 + 1] / 0.0029f;
            float q2 =  params[m * 3 + 2] / 0.0018f;
            float d0v = (q0 - T0) / 0.25f, d1v = (q1 - T1) / 0.25f, d2v = (q2 - T2) / 0.25f;
            float dist2 = d0v * d0v + d1v * d1v + d2v * d2v;
            sw[m] = expf(-dist2 / 2.0f);
            ssw += sw[m];
        }
        float wsum = 0.0f, wv[MM];
        for (int m = 0; m < MM; ++m) { wv[m] = attn_w[m] * (sw[m] / ssw); wsum += wv[m]; }
        for (int m = 0; m < MM; ++m) w_out[m] = wv[m] / wsum;
    }
}

// ---------------------------------------------------------------------------
// Kernel B: per-(member, output-column) interpolation tables.
//   Replicates jnp.searchsorted(ys, yq, 'right')-1 on the member grid,
//   including the out-of-bounds zero-fill at yq=75 when lys*(75/lys) < 75.
//   Weights are pre-multiplied by the blend weight w[m].
// ---------------------------------------------------------------------------
__global__ __launch_bounds__(256)
void tables_kernel(const float* __restrict__ params, const float* __restrict__ w,
                   int* __restrict__ iyT, float* __restrict__ waT,
                   float* __restrict__ wbT) {
    const int m = blockIdx.x;
    const int j = threadIdx.x;
    const float lys = params[m * 3 + 0];
    const float s   = 75.0f / lys;
    const float yq  = (j == 255) ? 75.0f : (float)j * (75.0f / 255.0f);

    // searchsorted right: lo = count of {k : ys(k) <= yq}
    int lo = 0, hiB = 256;
    while (lo < hiB) {
        int mid = (lo + hiB) >> 1;
        if (ys_val(mid, lys, s) <= yq) lo = mid + 1; else hiB = mid;
    }
    int iy = lo - 1;
    iy = iy < 0 ? 0 : (iy > 254 ? 254 : iy);

    const float y0 = ys_val(iy, lys, s);
    const float y1 = ys_val(iy + 1, lys, s);
    const float ty = (yq - y0) / (y1 - y0);
    const bool inb = (yq <= ys_val(255, lys, s));   // lower bound ys(0)=0 always ok

    const float wm = w[m];
    const int k = m * NG + j;
    iyT[k] = iy;
    waT[k] = inb ? wm * (1.0f - ty) : 0.0f;
    wbT[k] = inb ? wm * ty          : 0.0f;
}

// ---------------------------------------------------------------------------
// Kernel C: the streaming weighted gather (bandwidth bound, ~340 MB total).
//   block = 256 threads (one output column each), handles one t and 32 rows i.
//   Per-thread (col, wa, wb) for all 12 members live in registers; source
//   loads are coalesced since iy[j] ~= j.
// ---------------------------------------------------------------------------
__global__ __launch_bounds__(256)
void combine_kernel(const float* __restrict__ c1, const float* __restrict__ c2,
                    const int* __restrict__ iyT, const float* __restrict__ waT,
                    const float* __restrict__ wbT, float* __restrict__ out) {
    __shared__ int   iyS[MM * NG];
    __shared__ float waS[MM * NG];
    __shared__ float wbS[MM * NG];
    const int tid = threadIdx.x;
    for (int k = tid; k < MM * NG; k += 256) {
        iyS[k] = iyT[k]; waS[k] = waT[k]; wbS[k] = wbT[k];
    }
    __syncthreads();

    const int t  = blockIdx.y;
    const int i0 = blockIdx.x * 32;

    // tidx[t] = clip(int(times[t]/200*64), 0, 63), times = linspace(0,200,50)
    float tt = (t == 49) ? 200.0f : (float)t * (200.0f / 49.0f);
    int ti = (int)((tt / 200.0f) * 64.0f);
    ti = ti < 0 ? 0 : (ti > 63 ? 63 : ti);

    const int j = tid;
    int   off[MM]; float wa[MM], wb[MM];
#pragma unroll
    for (int m = 0; m < MM; ++m) {
        int k = m * NG + j;
        off[m] = ((m * TSOL + ti) << 16) + iyS[k];   // (m*64+ti)*256*256 + col
        wa[m]  = waS[k];
        wb[m]  = wbS[k];
    }

    for (int ii = 0; ii < 32; ++ii) {
        const int i = i0 + ii;
        const int rowoff = i << 8;
        float a1 = 0.0f, a2 = 0.0f;
#pragma unroll
        for (int m = 0; m < MM; ++m) {
            const int idx = off[m] + rowoff;
            a1 += wa[m] * c1[idx] + wb[m] * c1[idx + 1];
            a2 += wa[m] * c2[idx] + wb[m] * c2[idx + 1];
        }
        if (j == 0)   a1 = 0.001f;    // c1[:, :, 0]  = C_CU_TARGET
        if (j == 255) a2 = 0.0001f;   // c2[:, :, -1] = C_NI_TARGET
        out[((0 * NT + t) * NG + i) * NG + j] = a1;
        out[((1 * NT + t) * NG + i) * NG + j] = a2;
    }
}

// ---------------------------------------------------------------------------
extern "C" void kernel_launch(void* const* d_in, const int* in_sizes, int n_in,
                              void* d_out, int out_size, void* d_ws, size_t ws_size,
                              hipStream_t stream) {
    const float* params = (const float*)d_in[0];
    const float* c1     = (const float*)d_in[1];
    const float* c2     = (const float*)d_in[2];
    const float* Wq     = (const float*)d_in[3];
    const float* bq     = (const float*)d_in[4];
    const float* Wk     = (const float*)d_in[5];
    const float* bk     = (const float*)d_in[6];
    float* out = (float*)d_out;

    char*  ws  = (char*)d_ws;
    float* w   = (float*)ws;                              // 12 floats
    int*   iyT = (int*)  (ws + 64);                       // 12*256 ints
    float* waT = (float*)(ws + 64 + MM * NG * 4);         // 12*256 floats
    float* wbT = (float*)(ws + 64 + 2 * MM * NG * 4);     // 12*256 floats

    weights_kernel<<<1, 32, 0, stream>>>(params, Wq, bq, Wk, bk, w);
    tables_kernel<<<MM, 256, 0, stream>>>(params, w, iyT, waT, wbT);
    dim3 grid(NG / 32, NT);
    combine_kernel<<<grid, 256, 0, stream>>>(c1, c2, iyT, waT, wbT, out);
}